// FFT_31155692765574
// MI455X (gfx1250) — compile-verified
//
#include <hip/hip_runtime.h>
#include <math.h>

// ---------------------------------------------------------------------------
// |FFT(x)| of N = 2^24 f32 reals on gfx1250 (MI455X).
//
// Bailey four-step: N = 4096 x 4096.
//   Pass 1: per column n2, 4096-pt FFT over n1; multiply W_N^{k1*n2};
//           store complex to ws in [k1][n2] layout (pass-2 rows contiguous).
//   Pass 2: per row k1, 4096-pt FFT over n2; out[k1 + 4096*k2] = |D[k1][k2]|.
//
// Each 4096-pt FFT = three radix-16 stages; each stage is 16 batches of
// "DFT16 x (16 vectors)" = complex 16x16x16 matmuls executed with
// V_WMMA_F32_16X16X4_F32 (4 chained K=4 WMMAs per real matmul).
// DFT-16 matrix + twiddles generated in-register with __sincosf.
// LDS tile is a static __shared__ array so all tile traffic lowers to DS ops
// (not FLAT generic ops). Needs ws_size >= 2^24 * 8 B = 128 MB.
// ---------------------------------------------------------------------------

typedef float v2f __attribute__((ext_vector_type(2)));
typedef float v8f __attribute__((ext_vector_type(8)));

#define NSIDE      4096
#define VEC_PER_WG 4
#define THREADS    256

__device__ __forceinline__ v8f wmma_f32(v2f a, v2f b, v8f c) {
  // D = A(16x4) * B(4x16) + C ; f32 WMMA, 8-arg form:
  // (neg_a, A, neg_b, B, c_mod, C, reuse_a, reuse_b)
  return __builtin_amdgcn_wmma_f32_16x16x4_f32(false, a, false, b,
                                               (short)0, c, false, false);
}

// In-place 4096-pt complex FFT of v[0..4095], natural-order output.
// All 256 threads of the block participate. Index split n = n0+16*n1+256*n2,
// output k = k0+16*k1+256*k2:
//   stage0: DFT16 over n2 (addr = (16b+nl) + 256*K), store * W_256^{b*M}
//   stage1: DFT16 over n1 (addr = nl + 16*K + 256*b), store * W_4096^{nl*(b+16M)}
//   stage2: DFT16 over n0 (addr = K + 16*nl + 256*b), store at b+16*nl+256*M
__device__ __forceinline__ void fft4096(float2* v) {
  const int tid  = threadIdx.x;
  const int lane = tid & 31;
  const int wave = tid >> 5;
  const int nl   = lane & 15;   // A: row M ; B: col N ; C/D: col N
  const int hi   = lane >> 4;   // lane-half selects K/M sub-range

  // Constant DFT-16 matrix as WMMA A operand (reused by all stages/batches).
  v2f a_re[4], a_im[4], a_imn[4];
#pragma unroll
  for (int kc = 0; kc < 4; ++kc) {
    const int K0 = kc * 4 + hi * 2;
    float s0, c0, s1, c1;
    __sincosf(-0.39269908169872414f * (float)((nl * K0) & 15), &s0, &c0);
    __sincosf(-0.39269908169872414f * (float)((nl * (K0 + 1)) & 15), &s1, &c1);
    a_re[kc].x = c0;  a_re[kc].y = c1;
    a_im[kc].x = s0;  a_im[kc].y = s1;
    a_imn[kc].x = -s0; a_imn[kc].y = -s1;
  }

#pragma unroll
  for (int stage = 0; stage < 3; ++stage) {
    v2f b_re[2][4], b_im[2][4];
    // Load both of this wave's batches fully before anyone stores (stage 2's
    // digit permutation crosses batch address sets).
#pragma unroll
    for (int rep = 0; rep < 2; ++rep) {
      const int b = wave * 2 + rep;
#pragma unroll
      for (int kc = 0; kc < 4; ++kc) {
#pragma unroll
        for (int w = 0; w < 2; ++w) {
          const int K = kc * 4 + hi * 2 + w;
          int addr;
          if (stage == 0)      addr = (b * 16 + nl) + 256 * K;
          else if (stage == 1) addr = nl + 16 * K + 256 * b;
          else                 addr = K + 16 * nl + 256 * b;
          const float2 t = v[addr];
          if (w == 0) { b_re[rep][kc].x = t.x; b_im[rep][kc].x = t.y; }
          else        { b_re[rep][kc].y = t.x; b_im[rep][kc].y = t.y; }
        }
      }
    }
    __syncthreads();

#pragma unroll
    for (int rep = 0; rep < 2; ++rep) {
      const int b = wave * 2 + rep;
      v8f acc_re = {};
      v8f acc_im = {};
      // C_re = Fre*Bre - Fim*Bim ; C_im = Fre*Bim + Fim*Bre
#pragma unroll
      for (int kc = 0; kc < 4; ++kc) acc_re = wmma_f32(a_re[kc],  b_re[rep][kc], acc_re);
#pragma unroll
      for (int kc = 0; kc < 4; ++kc) acc_re = wmma_f32(a_imn[kc], b_im[rep][kc], acc_re);
#pragma unroll
      for (int kc = 0; kc < 4; ++kc) acc_im = wmma_f32(a_re[kc],  b_im[rep][kc], acc_im);
#pragma unroll
      for (int kc = 0; kc < 4; ++kc) acc_im = wmma_f32(a_im[kc],  b_re[rep][kc], acc_im);

      // Twiddle + scatter back to LDS.
#pragma unroll
      for (int vv = 0; vv < 8; ++vv) {
        const int M = vv + 8 * hi;
        float cr = acc_re[vv], ci = acc_im[vv];
        int addr;
        if (stage == 0) {
          float s, c;
          __sincosf(-0.02454369260617026f * (float)(b * M), &s, &c); // 2pi/256
          const float nr = cr * c - ci * s, ni = cr * s + ci * c;
          cr = nr; ci = ni;
          addr = (b * 16 + nl) + 256 * M;
        } else if (stage == 1) {
          float s, c;
          __sincosf(-0.001533980787885641f * (float)(nl * (b + 16 * M)), &s, &c); // 2pi/4096
          const float nr = cr * c - ci * s, ni = cr * s + ci * c;
          cr = nr; ci = ni;
          addr = nl + 16 * M + 256 * b;
        } else {
          addr = b + 16 * nl + 256 * M;  // natural-order output
        }
        v[addr] = make_float2(cr, ci);
      }
    }
    __syncthreads();
  }
}

// Pass 1: columns n2 = 4g..4g+3; FFT over n1; twiddle; ws[k1*4096 + n2].
__global__ void __launch_bounds__(THREADS)
fft_pass1(const float* __restrict__ x, float2* __restrict__ ws) {
  __shared__ float2 lds[VEC_PER_WG * NSIDE];  // 128 KB static LDS
  const int n2base = blockIdx.x * VEC_PER_WG;
  const int tid = threadIdx.x;

  // Gather 4 columns: threads t..t+3 read 16 contiguous bytes of one row.
  for (int e = tid; e < VEC_PER_WG * NSIDE; e += THREADS) {
    const int c = e & 3, n1 = e >> 2;
    lds[c * NSIDE + n1] = make_float2(x[(size_t)n1 * NSIDE + n2base + c], 0.0f);
  }
  __syncthreads();

#pragma unroll 1
  for (int c = 0; c < VEC_PER_WG; ++c) fft4096(lds + c * NSIDE);

  // Inter-pass twiddle W_N^{k1*n2}, store transposed: ws[k1][n2].
  for (int e = tid; e < VEC_PER_WG * NSIDE; e += THREADS) {
    const int c = e & 3, k1 = e >> 2;
    const float2 val = lds[c * NSIDE + k1];
    const unsigned m = (unsigned)k1 * (unsigned)(n2base + c);  // < 2^24, exact in f32
    float s, cth;
    __sincosf(-3.74485598918757e-7f * (float)m, &s, &cth);  // -2*pi*m / 2^24
    const float2 r = make_float2(val.x * cth - val.y * s, val.x * s + val.y * cth);
    ws[(size_t)k1 * NSIDE + (n2base + c)] = r;
  }
}

// Pass 2: rows k1 = 4g..4g+3 of ws (contiguous); FFT over n2; |.| out.
__global__ void __launch_bounds__(THREADS)
fft_pass2(const float2* __restrict__ ws, float* __restrict__ out) {
  __shared__ float2 lds[VEC_PER_WG * NSIDE];  // 128 KB static LDS
  const int k1base = blockIdx.x * VEC_PER_WG;
  const int tid = threadIdx.x;

  // Fully coalesced row loads (256 B per wave per iteration).
  for (int e = tid; e < VEC_PER_WG * NSIDE; e += THREADS) {
    const int c = e >> 12, n2 = e & (NSIDE - 1);
    lds[c * NSIDE + n2] = ws[(size_t)(k1base + c) * NSIDE + n2];
  }
  __syncthreads();

#pragma unroll 1
  for (int c = 0; c < VEC_PER_WG; ++c) fft4096(lds + c * NSIDE);

  // out[k1 + 4096*k2] = |D[k1][k2]|; threads t..t+3 write 16 contiguous bytes.
  for (int e = tid; e < VEC_PER_WG * NSIDE; e += THREADS) {
    const int c = e & 3, k2 = e >> 2;
    const float2 d = lds[c * NSIDE + k2];
    out[(size_t)(k1base + c) + (size_t)NSIDE * k2] = sqrtf(d.x * d.x + d.y * d.y);
  }
}

extern "C" void kernel_launch(void* const* d_in, const int* in_sizes, int n_in,
                              void* d_out, int out_size, void* d_ws, size_t ws_size,
                              hipStream_t stream) {
  const float* x = (const float*)d_in[0];
  float* out = (float*)d_out;
  float2* ws = (float2*)d_ws;  // needs 2^24 * 8 B = 128 MB

  const dim3 grid(NSIDE / VEC_PER_WG), block(THREADS);
  fft_pass1<<<grid, block, 0, stream>>>(x, ws);
  fft_pass2<<<grid, block, 0, stream>>>(ws, out);
}